// TopologicalDecoder_11407433138226
// MI455X (gfx1250) — compile-verified
//
#include <hip/hip_runtime.h>
#include <math.h>

// ---------------------------------------------------------------------------
// Problem constants (match reference)
// ---------------------------------------------------------------------------
#define B_   2
#define T_   2048
#define D_   1024
#define V_   32000
#define NF_  4

typedef __attribute__((ext_vector_type(2))) float v2f;
typedef __attribute__((ext_vector_type(4))) float v4f;
typedef __attribute__((ext_vector_type(8))) float v8f;

// Epilogue modes
#define EPI_NONE 0
#define EPI_BIAS 1
#define EPI_GELU 2   // bias + exact GELU
#define EPI_ADJ  3   // sigmoid(10*(c - thresh))
#define EPI_ACC2 4   // u = 0.2*(c+bias); C += u; D2 += u

__device__ __forceinline__ float gelu_exact(float x) {
    return 0.5f * x * (1.0f + erff(x * 0.70710678118654752440f));
}
__device__ __forceinline__ float sigmoidf_(float x) {
    return 1.0f / (1.0f + expf(-x));
}

// Async global->LDS copy of 16 bytes per lane (CDNA5 ASYNCcnt path).
__device__ __forceinline__ void async_copy_b128(unsigned lds_off, const float* gptr) {
    asm volatile("global_load_async_to_lds_b128 %0, %1, off"
                 :: "v"(lds_off), "v"(gptr) : "memory");
}
__device__ __forceinline__ void wait_async0() {
    asm volatile("s_wait_asynccnt 0x0" ::: "memory");
}

// ---------------------------------------------------------------------------
// fp32 WMMA GEMM: C[M,N] = epi(A[M,K] * B) using V_WMMA_F32_16X16X4_F32.
// BT=false: B is row-major K x N (ldb = row stride).
// BT=true : B is row-major N x K (computes A * B^T), used for xn @ xn^T.
// Block tile 128x128, TK=16, 8 waves (wave32) arranged 4(M) x 2(N),
// 32x64 per wave (2x4 accumulators). LDS double-buffered; A tile and
// transposed-B tile are filled with global_load_async_to_lds_b128 and
// synchronized with s_wait_asynccnt; K x N weights are transposed into LDS
// so every WMMA fragment is one aligned ds_load_b64.
// ---------------------------------------------------------------------------
template <int EPI, bool BT>
__global__ __launch_bounds__(256) void gemm_wmma_f32(
    const float* __restrict__ A, int lda,
    const float* __restrict__ Bm, int ldb,
    float* __restrict__ C, int ldc,
    int M, int N, int K,
    const float* __restrict__ bias,
    const float* __restrict__ thresh_ptr,
    float* __restrict__ D2)
{
    constexpr int TM = 128, TN = 128, TK = 16, LDK = TK + 4; // 20 floats = 80B rows (16B aligned)
    __shared__ float As[2][TM][LDK];   // [m][k]
    __shared__ float Bs[2][TN][LDK];   // [n][k]  (transposed: fragment = contiguous b64)

    const int tid   = threadIdx.x;
    const int wave  = tid >> 5;
    const int lane  = tid & 31;
    const int laneN = lane & 15;        // M of A-frag / N of B-frag / N of C
    const int laneH = lane >> 4;        // half-wave select (K pair for A/B, +8 rows for C)
    const int wm    = (wave & 3) << 5;  // 0,32,64,96
    const int wn    = (wave >> 2) << 6; // 0,64
    const int bm    = blockIdx.y * TM;
    const int bn    = blockIdx.x * TN;

    float th = 0.0f;
    if constexpr (EPI == EPI_ADJ) th = *thresh_ptr;   // uniform scalar load

    v8f acc[2][4];
#pragma unroll
    for (int i = 0; i < 2; ++i)
#pragma unroll
        for (int j = 0; j < 4; ++j)
#pragma unroll
            for (int e = 0; e < 8; ++e) acc[i][j][e] = 0.0f;

    // Issue the global->LDS copies for K-slab k0 into buffer `buf`.
    auto loadTiles = [&](int k0, int buf) {
        // A tile: 128 x 16 floats = 512 x b128, 2 per thread, async.
#pragma unroll
        for (int i = 0; i < 2; ++i) {
            int idx = tid + i * 256;
            int r = idx >> 2;             // 4 x b128 per row
            int c = (idx & 3) << 2;
            const float* gp = A + (size_t)(bm + r) * lda + (k0 + c);
            async_copy_b128((unsigned)(size_t)&As[buf][r][c], gp);
        }
        if constexpr (BT) {
            // B stored N x K: rows copy straight into Bs[n][k], async.
#pragma unroll
            for (int i = 0; i < 2; ++i) {
                int idx = tid + i * 256;
                int n = idx >> 2;
                int c = (idx & 3) << 2;
                const float* gp = Bm + (size_t)(bn + n) * ldb + (k0 + c);
                async_copy_b128((unsigned)(size_t)&Bs[buf][n][c], gp);
            }
        } else {
            // B stored K x N: load b128 along N, transpose into Bs[n][k].
#pragma unroll
            for (int i = 0; i < 2; ++i) {
                int idx = tid + i * 256;
                int kk = idx >> 5;            // 16 k-rows, 32 x b128 each
                int c = (idx & 31) << 2;
                v4f v = *(const v4f*)(Bm + (size_t)(k0 + kk) * ldb + (bn + c));
                Bs[buf][c + 0][kk] = v.x;
                Bs[buf][c + 1][kk] = v.y;
                Bs[buf][c + 2][kk] = v.z;
                Bs[buf][c + 3][kk] = v.w;
            }
        }
    };

    loadTiles(0, 0);
    int cur = 0;
    for (int k0 = 0; k0 < K; k0 += TK) {
        wait_async0();        // own async copies into buf `cur` complete
        __syncthreads();      // all waves' copies visible; prev reads of buf cur^1 done
        if (k0 + TK < K) loadTiles(k0 + TK, cur ^ 1);   // overlap next slab with compute

        // 4 K-steps of 4, 8 WMMAs each; every fragment is one ds_load_b64.
#pragma unroll
        for (int kk = 0; kk < TK; kk += 4) {
            const int kA = kk + (laneH << 1);
            const int ar = wm + laneN;
            v2f a0 = *(const v2f*)&As[cur][ar][kA];
            v2f a1 = *(const v2f*)&As[cur][ar + 16][kA];
#pragma unroll
            for (int j = 0; j < 4; ++j) {
                v2f b = *(const v2f*)&Bs[cur][wn + (j << 4) + laneN][kA];
                acc[0][j] = __builtin_amdgcn_wmma_f32_16x16x4_f32(false, a0, false, b, (short)0, acc[0][j], false, false);
                acc[1][j] = __builtin_amdgcn_wmma_f32_16x16x4_f32(false, a1, false, b, (short)0, acc[1][j], false, false);
            }
        }
        cur ^= 1;
    }

    // ---- epilogue: C/D layout -> VGPR r holds M = r (lanes 0-15) / r+8 (16-31) ----
#pragma unroll
    for (int i = 0; i < 2; ++i) {
#pragma unroll
        for (int j = 0; j < 4; ++j) {
            const int row0 = bm + wm + (i << 4) + (laneH << 3);
            const int col  = bn + wn + (j << 4) + laneN;
#pragma unroll
            for (int r = 0; r < 8; ++r) {
                float c = acc[i][j][r];
                const size_t idx = (size_t)(row0 + r) * (size_t)ldc + (size_t)col;
                if constexpr (EPI == EPI_NONE) {
                    C[idx] = c;
                } else if constexpr (EPI == EPI_BIAS) {
                    C[idx] = c + bias[col];
                } else if constexpr (EPI == EPI_GELU) {
                    C[idx] = gelu_exact(c + bias[col]);
                } else if constexpr (EPI == EPI_ADJ) {
                    C[idx] = sigmoidf_(10.0f * (c - th));
                } else { // EPI_ACC2
                    float u = (c + bias[col]) * 0.2f;
                    C[idx]  += u;
                    D2[idx] += u;
                }
            }
        }
    }
}

// ---------------------------------------------------------------------------
// x[bt,:] = embed[tokens[bt],:] + pos_embed[bt % T,:]
// ---------------------------------------------------------------------------
__global__ __launch_bounds__(256) void embed_kernel(
    const int* __restrict__ tok, const float* __restrict__ emb,
    const float* __restrict__ pos, float* __restrict__ x)
{
    const int bt = blockIdx.x;
    const int t  = bt & (T_ - 1);
    const int tk = tok[bt];
    const float* e = emb + (size_t)tk * D_;
    const float* p = pos + (size_t)t * D_;
    float* o = x + (size_t)bt * D_;
    for (int d = threadIdx.x; d < D_; d += 256) o[d] = e[d] + p[d];
}

// ---------------------------------------------------------------------------
// thresholds = sigmoid(gelu(mean(c_states) @ thr_W1 + b1) @ thr_W2 + b2)
// smallbuf[0..3] = thresholds, smallbuf[4] = 0.5 (betti threshold)
// ---------------------------------------------------------------------------
__global__ __launch_bounds__(256) void thr_kernel(
    const float* __restrict__ cs, const float* __restrict__ W1,
    const float* __restrict__ b1, const float* __restrict__ W2,
    const float* __restrict__ b2, float* __restrict__ smallbuf)
{
    __shared__ float cp[D_];
    __shared__ float h[32];
    const int tid = threadIdx.x;
    for (int d = tid; d < D_; d += 256) {
        float s = 0.0f;
        for (int r = 0; r < 8; ++r) s += cs[r * D_ + d];
        cp[d] = s * 0.125f;
    }
    __syncthreads();
    if (tid < 32) {
        float s = b1[tid];
        for (int k = 0; k < D_; ++k) s += cp[k] * W1[k * 32 + tid];
        h[tid] = gelu_exact(s);
    }
    __syncthreads();
    if (tid < NF_) {
        float s = b2[tid];
        for (int k = 0; k < 32; ++k) s += h[k] * W2[k * NF_ + tid];
        smallbuf[tid] = sigmoidf_(s);
    }
    if (tid == NF_) smallbuf[NF_] = 0.5f;
}

// ---------------------------------------------------------------------------
// Row L2-normalize x0 -> xn; optionally copy x0 into left half of combined.
// ---------------------------------------------------------------------------
__global__ __launch_bounds__(256) void rownorm_kernel(
    const float* __restrict__ x0, float* __restrict__ xn, float* __restrict__ comb_left)
{
    __shared__ float red[256];
    const int row = blockIdx.x;
    const float* src = x0 + (size_t)row * D_;
    float s = 0.0f;
    for (int d = threadIdx.x; d < D_; d += 256) { float v = src[d]; s += v * v; }
    red[threadIdx.x] = s;
    __syncthreads();
    for (int off = 128; off > 0; off >>= 1) {
        if (threadIdx.x < off) red[threadIdx.x] += red[threadIdx.x + off];
        __syncthreads();
    }
    const float inv = 1.0f / fmaxf(sqrtf(red[0]), 1e-12f);
    float* dn = xn + (size_t)row * D_;
    float* dc = comb_left ? comb_left + (size_t)row * (2 * D_) : nullptr;
    for (int d = threadIdx.x; d < D_; d += 256) {
        float v = src[d];
        dn[d] = v * inv;
        if (dc) dc[d] = v;
    }
}

// degree[i] = sum_j adj[i,j]
__global__ __launch_bounds__(256) void degree_kernel(
    const float* __restrict__ adj, float* __restrict__ degree)
{
    __shared__ float red[256];
    const int row = blockIdx.x;
    const float* src = adj + (size_t)row * T_;
    float s = 0.0f;
    for (int j = threadIdx.x; j < T_; j += 256) s += src[j];
    red[threadIdx.x] = s;
    __syncthreads();
    for (int off = 128; off > 0; off >>= 1) {
        if (threadIdx.x < off) red[threadIdx.x] += red[threadIdx.x + off];
        __syncthreads();
    }
    if (threadIdx.x == 0) degree[row] = red[0];
}

// beta0 = count(degree < 0.5); beta1 = relu(sum(degree)/2 - T + beta0 + 1)
__global__ __launch_bounds__(256) void beta_kernel(
    const float* __restrict__ degree, float* __restrict__ smallbuf)
{
    __shared__ float scnt[256];
    __shared__ float ssum[256];
    float cnt = 0.0f, sum = 0.0f;
    for (int i = threadIdx.x; i < T_; i += 256) {
        float d = degree[i];
        sum += d;
        if (d < 0.5f) cnt += 1.0f;
    }
    scnt[threadIdx.x] = cnt; ssum[threadIdx.x] = sum;
    __syncthreads();
    for (int off = 128; off > 0; off >>= 1) {
        if (threadIdx.x < off) {
            scnt[threadIdx.x] += scnt[threadIdx.x + off];
            ssum[threadIdx.x] += ssum[threadIdx.x + off];
        }
        __syncthreads();
    }
    if (threadIdx.x == 0) {
        float beta0 = scnt[0];
        float n_edges = ssum[0] * 0.5f;
        float beta1 = fmaxf(0.0f, n_edges - (float)T_ + beta0 + 1.0f);
        smallbuf[5] = beta0;
        smallbuf[6] = beta1;
    }
}

// b1eff[j] = int_b1[j] + beta0*int_W1[1024,j] + beta1*int_W1[1025,j]
// (Betti features are constant across rows -> fold the K=1026 remainder into bias.)
__global__ __launch_bounds__(256) void b1eff_kernel(
    const float* __restrict__ intW1, const float* __restrict__ intb1,
    const float* __restrict__ smallbuf, float* __restrict__ b1eff)
{
    const int j = blockIdx.x * 256 + threadIdx.x;
    if (j >= 2 * D_) return;
    const float beta0 = smallbuf[5], beta1 = smallbuf[6];
    b1eff[j] = intb1[j]
             + beta0 * intW1[(size_t)D_       * (2 * D_) + j]
             + beta1 * intW1[(size_t)(D_ + 1) * (2 * D_) + j];
}

// In-place LayerNorm over last dim (1024) with gain/bias.
__global__ __launch_bounds__(256) void ln_kernel(
    float* __restrict__ y, const float* __restrict__ g, const float* __restrict__ b)
{
    __shared__ float s1[256];
    __shared__ float s2[256];
    const int row = blockIdx.x;
    float* p = y + (size_t)row * D_;
    float a = 0.0f, q = 0.0f;
    for (int d = threadIdx.x; d < D_; d += 256) { float v = p[d]; a += v; q += v * v; }
    s1[threadIdx.x] = a; s2[threadIdx.x] = q;
    __syncthreads();
    for (int off = 128; off > 0; off >>= 1) {
        if (threadIdx.x < off) {
            s1[threadIdx.x] += s1[threadIdx.x + off];
            s2[threadIdx.x] += s2[threadIdx.x + off];
        }
        __syncthreads();
    }
    const float mean = s1[0] * (1.0f / D_);
    const float var  = s2[0] * (1.0f / D_) - mean * mean;
    const float rstd = rsqrtf(var + 1e-5f);
    for (int d = threadIdx.x; d < D_; d += 256)
        p[d] = (p[d] - mean) * rstd * g[d] + b[d];
}

// ---------------------------------------------------------------------------
// Host-side orchestration
// ---------------------------------------------------------------------------
extern "C" void kernel_launch(void* const* d_in, const int* in_sizes, int n_in,
                              void* d_out, int out_size, void* d_ws, size_t ws_size,
                              hipStream_t stream)
{
    (void)in_sizes; (void)n_in; (void)out_size; (void)ws_size;

    const int*   tokens   = (const int*)  d_in[0];
    const float* c_states = (const float*)d_in[1];
    const float* embedW   = (const float*)d_in[2];
    const float* posW     = (const float*)d_in[3];
    // d_in[4..7] = topo_* (dead code in forward)
    const float* thr_W1   = (const float*)d_in[8];
    const float* thr_b1   = (const float*)d_in[9];
    const float* thr_W2   = (const float*)d_in[10];
    const float* thr_b2   = (const float*)d_in[11];
    const float* simp_W1  = (const float*)d_in[12];
    const float* simp_b1  = (const float*)d_in[13];
    const float* simp_W2  = (const float*)d_in[14];
    const float* simp_b2  = (const float*)d_in[15];
    const float* int_W1   = (const float*)d_in[16];
    const float* int_b1   = (const float*)d_in[17];
    const float* int_W2   = (const float*)d_in[18];
    const float* int_b2   = (const float*)d_in[19];
    const float* ln_g     = (const float*)d_in[20];
    const float* ln_b     = (const float*)d_in[21];
    const float* head_W   = (const float*)d_in[22];

    float* out = (float*)d_out;
    float* ws  = (float*)d_ws;

    // Workspace layout (floats). int-MLP buffers alias the level-phase buffers.
    const size_t off_x    = 0;                         // [B*T, D]   4M
    const size_t off_xn   = off_x   + (size_t)B_*T_*D_;// [T, D]     2M
    const size_t off_adj  = off_xn  + (size_t)T_*D_;   // [T, T]     4M
    const size_t off_comb = off_adj + (size_t)T_*T_;   // [T, 2D]    4M
    const size_t off_h    = off_comb+ (size_t)T_*2*D_; // [T, D]     2M
    const size_t off_h2   = off_adj;                   // [B*T, 2D]  8M (aliases adj+comb, used after betti)
    const size_t off_y    = off_h;                     // [B*T, D]   4M (aliases h and beyond)
    const size_t off_sm   = off_h + (size_t)B_*T_*D_;  // small scalars
    const size_t off_deg  = off_sm + 16;               // [T]
    const size_t off_b1e  = off_deg + T_;              // [2D]

    float* x     = ws + off_x;
    float* x1    = x + (size_t)T_ * D_;                // batch element 1
    float* xn    = ws + off_xn;
    float* adj   = ws + off_adj;
    float* comb  = ws + off_comb;
    float* h     = ws + off_h;
    float* h2    = ws + off_h2;
    float* y     = ws + off_y;
    float* smallb= ws + off_sm;
    float* deg   = ws + off_deg;
    float* b1eff = ws + off_b1e;

    const dim3 blk(256);

    // 1) x = embed[tokens] + pos_embed
    embed_kernel<<<dim3(B_ * T_), blk, 0, stream>>>(tokens, embedW, posW, x);

    // 2) thresholds (+ 0.5 constant at smallb[4])
    thr_kernel<<<dim3(1), blk, 0, stream>>>(c_states, thr_W1, thr_b1, thr_W2, thr_b2, smallb);

    // 3) simplex levels
    for (int l = 0; l < NF_; ++l) {
        // xn = l2norm(x0); comb[:, :D] = x0
        rownorm_kernel<<<dim3(T_), blk, 0, stream>>>(x, xn, comb);

        // adj = sigmoid(10*(xn @ xn^T - thr[l]))   [T,T]
        gemm_wmma_f32<EPI_ADJ, true><<<dim3(T_ / 128, T_ / 128), blk, 0, stream>>>(
            xn, D_, xn, D_, adj, T_, T_, T_, D_, nullptr, smallb + l, nullptr);

        // comb[:, D:2D] = adj @ x0   [T,D]
        gemm_wmma_f32<EPI_NONE, false><<<dim3(D_ / 128, T_ / 128), blk, 0, stream>>>(
            adj, T_, x, D_, comb + D_, 2 * D_, T_, D_, T_, nullptr, nullptr, nullptr);

        // h = gelu(comb @ simp_W1[l] + simp_b1[l])   [T,D]
        gemm_wmma_f32<EPI_GELU, false><<<dim3(D_ / 128, T_ / 128), blk, 0, stream>>>(
            comb, 2 * D_, simp_W1 + (size_t)l * 2 * D_ * D_, D_, h, D_,
            T_, D_, 2 * D_, simp_b1 + (size_t)l * D_, nullptr, nullptr);

        // update = h @ simp_W2[l] + simp_b2[l]; x0 += 0.2*update; x1 += 0.2*update
        gemm_wmma_f32<EPI_ACC2, false><<<dim3(D_ / 128, T_ / 128), blk, 0, stream>>>(
            h, D_, simp_W2 + (size_t)l * D_ * D_, D_, x, D_,
            T_, D_, D_, simp_b2 + (size_t)l * D_, nullptr, x1);
    }

    // 4) Betti numbers on final x0 (thresh = 0.5 at smallb[4])
    rownorm_kernel<<<dim3(T_), blk, 0, stream>>>(x, xn, nullptr);
    gemm_wmma_f32<EPI_ADJ, true><<<dim3(T_ / 128, T_ / 128), blk, 0, stream>>>(
        xn, D_, xn, D_, adj, T_, T_, T_, D_, nullptr, smallb + NF_, nullptr);
    degree_kernel<<<dim3(T_), blk, 0, stream>>>(adj, deg);
    beta_kernel<<<dim3(1), blk, 0, stream>>>(deg, smallb);

    // 5) fold betti rank-2 term into effective bias for integrate layer 1
    b1eff_kernel<<<dim3((2 * D_) / 256), blk, 0, stream>>>(int_W1, int_b1, smallb, b1eff);

    // 6) h2 = gelu(x @ int_W1[:D,:] + b1eff)   [B*T, 2D]   (aliases adj/comb; betti done)
    gemm_wmma_f32<EPI_GELU, false><<<dim3((2 * D_) / 128, (B_ * T_) / 128), blk, 0, stream>>>(
        x, D_, int_W1, 2 * D_, h2, 2 * D_, B_ * T_, 2 * D_, D_, b1eff, nullptr, nullptr);

    // 7) y = h2 @ int_W2 + int_b2   [B*T, D]
    gemm_wmma_f32<EPI_BIAS, false><<<dim3(D_ / 128, (B_ * T_) / 128), blk, 0, stream>>>(
        h2, 2 * D_, int_W2, D_, y, D_, B_ * T_, D_, 2 * D_, int_b2, nullptr, nullptr);

    // 8) LayerNorm in place
    ln_kernel<<<dim3(B_ * T_), blk, 0, stream>>>(y, ln_g, ln_b);

    // 9) logits = y @ head_W   [B*T, V] -> d_out
    gemm_wmma_f32<EPI_NONE, false><<<dim3(V_ / 128, (B_ * T_) / 128), blk, 0, stream>>>(
        y, D_, head_W, V_, out, V_, B_ * T_, V_, D_, nullptr, nullptr, nullptr);
}